// OrdinalRegressionLoss_69655779607159
// MI455X (gfx1250) — compile-verified
//
#include <hip/hip_runtime.h>

#define NUM_CLASSES 256

typedef __attribute__((ext_vector_type(2))) float v2f;
typedef __attribute__((ext_vector_type(8))) float v8f;

__global__ void OrdReg_zero_kernel(float* out) { out[0] = 0.0f; }

__device__ __forceinline__ float fmax0(float x) {
    // max(x, 0) in one op, no NaN-canonicalize pair
    return __builtin_amdgcn_fmed3f(x, 0.0f, __builtin_inff());
}

// One wave32 owns 16 rows. The entire row-wise cumsum runs on the matrix pipe:
//   d = A_chunk x U(upper-tri ones) + e   -> global inclusive prefix (4 WMMA, C = carry)
//   e += A_chunk x ones(16x16)            -> running total broadcast to all columns
// VALU does only the softplus; no cross-lane VALU/DS traffic, no carry adds.
__global__ __launch_bounds__(256) void OrdinalRegressionLoss_kernel(
    const float* __restrict__ logits,
    const int*   __restrict__ targets,
    float*       __restrict__ out,
    int numRows, float invB)
{
    const int lane        = threadIdx.x & 31;
    const int waveInBlk   = threadIdx.x >> 5;
    const int wavesPerBlk = blockDim.x >> 5;
    const long long waveId   = (long long)blockIdx.x * wavesPerBlk + waveInBlk;
    const long long numWaves = (long long)gridDim.x * wavesPerBlk;
    const int numTiles = numRows >> 4;          // 16 rows per tile

    const int n    = lane & 15;                 // output column within a 16-chunk
    const int hi   = (lane >> 4) & 1;           // lane half
    const int off2 = hi * 2;                    // A/B K sub-offset for 16x16x4 layout

    // B fragments: triangular U[k][n] = (k <= n) for the prefix, all-ones for totals.
    v2f tri[4], ones;
    #pragma unroll
    for (int j = 0; j < 4; ++j) {
        int k0 = 4 * j + off2;
        tri[j].x = (k0     <= n) ? 1.0f : 0.0f;
        tri[j].y = (k0 + 1 <= n) ? 1.0f : 0.0f;
    }
    ones.x = 1.0f; ones.y = 1.0f;

    float acc0 = 0.0f, acc1 = 0.0f;

    for (long long tile = waveId; tile < numTiles; tile += numWaves) {
        // Wave-uniform tile base -> SGPR, so targets ride the scalar path.
        const int rowBase = __builtin_amdgcn_readfirstlane((int)(tile << 4));

        // 16 targets for this tile via s_load; per-lane half select (one cndmask each).
        const int* tp = targets + rowBase;
        int tgt[8];
        #pragma unroll
        for (int i = 0; i < 8; ++i) {
            int tlo = tp[i];
            int thi = tp[i + 8];
            tgt[i] = hi ? thi : tlo;            // row m_i = rowBase + hi*8 + i
        }

        const float* rowPtr = logits + (size_t)(rowBase + n) * NUM_CLASSES;

        // Double-buffered A fragments: chunk c+1's loads fly during chunk c's compute.
        v2f a_nxt[4];
        #pragma unroll
        for (int j = 0; j < 4; ++j)
            a_nxt[j] = *(const v2f*)(rowPtr + 4 * j + off2);

        v8f e = {};                             // broadcast total of chunks < c
        #pragma unroll 4
        for (int c = 0; c < 16; ++c) {
            v2f a_cur[4];
            #pragma unroll
            for (int j = 0; j < 4; ++j) a_cur[j] = a_nxt[j];

            // Branchless wraparound load (last iter re-reads chunk 0, in-bounds)
            const float* p = rowPtr + ((c + 1) & 15) * 16 + off2;
            #pragma unroll
            for (int j = 0; j < 4; ++j)
                a_nxt[j] = *(const v2f*)(p + 4 * j);

            // Global inclusive prefix: carry enters through the C operand (no copy,
            // no VALU add) -- first WMMA reads e as C, writes d.
            v8f d = e;
            #pragma unroll
            for (int j = 0; j < 4; ++j)
                d = __builtin_amdgcn_wmma_f32_16x16x4_f32(
                        false, a_cur[j], false, tri[j],
                        (short)0, d, false, false);

            const int   k    = c * 16 + n;
            const float mask = (k == NUM_CLASSES - 1) ? 0.0f : 1.0f;
            #pragma unroll
            for (int i = 0; i < 8; ++i) {
                float cum = d[i];
                float x   = (k < tgt[i]) ? cum : -cum;        // -sign*cum ; |x| == |cum|
                // m first so -|m| folds into v_exp_f32's source modifiers
                float m   = cum * 1.4426950408889634f;
                float t   = __builtin_amdgcn_exp2f(-fabsf(m));
                float sp  = fmax0(x)
                          + 0.6931471805599453f * __builtin_amdgcn_logf(1.0f + t);
                if (i & 1) acc1 = fmaf(mask, sp, acc1);
                else       acc0 = fmaf(mask, sp, acc0);
            }

            // Advance the broadcast running total on the matrix pipe.
            #pragma unroll
            for (int j = 0; j < 4; ++j)
                e = __builtin_amdgcn_wmma_f32_16x16x4_f32(
                        false, a_cur[j], false, ones,
                        (short)0, e, false, false);
        }
    }

    float acc = acc0 + acc1;
    #pragma unroll
    for (int offs = 16; offs > 0; offs >>= 1)
        acc += __shfl_xor(acc, offs, 32);
    if (lane == 0)
        unsafeAtomicAdd(out, acc * invB);
}

extern "C" void kernel_launch(void* const* d_in, const int* in_sizes, int n_in,
                              void* d_out, int out_size, void* d_ws, size_t ws_size,
                              hipStream_t stream) {
    const float* logits  = (const float*)d_in[0];
    const int*   targets = (const int*)d_in[1];
    float*       out     = (float*)d_out;

    const int numRows = in_sizes[0] / NUM_CLASSES;   // 524288
    const float invB  = 1.0f / (float)numRows;

    OrdReg_zero_kernel<<<1, 1, 0, stream>>>(out);
    OrdinalRegressionLoss_kernel<<<2048, 256, 0, stream>>>(logits, targets, out,
                                                           numRows, invB);
}